// ExtendedQSUP_23519240913378
// MI455X (gfx1250) — compile-verified
//
#include <hip/hip_runtime.h>
#include <hip/hip_bf16.h>
#include <math.h>
#include <stdint.h>

// ---------------------------------------------------------------------------
// Problem constants (from reference)
// ---------------------------------------------------------------------------
#define BB   8192          // batch
#define DD   1024          // input dim
#define HH   2048          // hidden
#define SS   4             // streams
#define CC   1000          // classes
#define N1   (SS * 2 * HH) // 16384, GEMM1 N
#define TOPK_K 64
#define EPSF 1e-8f

typedef float v2f __attribute__((ext_vector_type(2)));
typedef float v8f __attribute__((ext_vector_type(8)));

// ---------------------------------------------------------------------------
// CDNA5 async global->LDS copy (8 bytes per lane, tracked by ASYNCcnt).
// LDS dest offset = low 32 bits of the generic pointer to __shared__.
// ---------------------------------------------------------------------------
__device__ __forceinline__ void async_copy_b64(float* ldsDst, const float* gSrc)
{
    unsigned lds          = (unsigned)(uintptr_t)ldsDst;
    unsigned long long ga = (unsigned long long)(uintptr_t)gSrc;
    asm volatile("global_load_async_to_lds_b64 %0, %1, off"
                 :: "v"(lds), "v"(ga) : "memory");
}

// ---------------------------------------------------------------------------
// Stage one 64x16 A tile + 128x16 B tile into LDS asynchronously.
// 1536 float2 chunks / 256 threads = 6 async ops per wave per stage.
// ---------------------------------------------------------------------------
template <int LDA, int LDB>
__device__ __forceinline__ void stage_tiles_async(
    const float* __restrict__ A, const float* __restrict__ Bm, int K,
    int m0, int n0, int k0, int tid,
    float* __restrict__ As, float* __restrict__ Bs)
{
    // A tile: 64 rows x 8 pairs = 512 chunks -> 2 per thread
    #pragma unroll
    for (int i = 0; i < 2; ++i) {
        const int c   = tid + i * 256;
        const int row = c >> 3;
        const int pr  = (c & 7) * 2;
        async_copy_b64(&As[row * LDA + pr],
                       A + (size_t)(m0 + row) * K + k0 + pr);
    }
    // B tile: 128 rows x 8 pairs = 1024 chunks -> 4 per thread
    #pragma unroll
    for (int i = 0; i < 4; ++i) {
        const int c   = tid + i * 256;
        const int row = c >> 3;
        const int pr  = (c & 7) * 2;
        async_copy_b64(&Bs[row * LDB + pr],
                       Bm + (size_t)(n0 + row) * K + k0 + pr);
    }
}

// ---------------------------------------------------------------------------
// WMMA fp32 NT GEMM:  C[m,n] = epi( sum_k A[m,k]*B[n,k] + bias[n] )
// A: M x K row-major, B: N x K row-major (computing A @ B^T).
// Block tile 64(M) x 128(N), BK=16, 256 threads = 8 waves (wave32).
// Each wave owns a 32x32 patch = 2x2 WMMA 16x16 tiles.
// Double-buffered LDS with async global->LDS copies (ASYNCcnt pipeline).
// EPI 0: gaussian exp(-v^2);  EPI 1: sigmoid
// ---------------------------------------------------------------------------
template <int EPI>
__global__ __launch_bounds__(256)
void qsup_gemm_nt_wmma(const float* __restrict__ A,
                       const float* __restrict__ Bm,
                       const float* __restrict__ bias,
                       float* __restrict__ C,
                       int M, int N, int K)
{
    constexpr int BM = 64, BN = 128, BK = 16;
    constexpr int LDA = BK + 2;   // 18 floats: 8B-aligned chunks, staggered banks
    constexpr int LDB = BK + 2;

    __shared__ float As[2][BM * LDA];
    __shared__ float Bs[2][BN * LDB];

    const int tid      = threadIdx.x;
    const int m0       = blockIdx.y * BM;
    const int n0       = blockIdx.x * BN;
    const int w        = tid >> 5;        // wave id 0..7
    const int lane     = tid & 31;
    const int wm       = w >> 2;          // 0..1  (M direction)
    const int wn       = w & 3;           // 0..3  (N direction)
    const int lane16   = lane & 15;
    const int laneHalf = lane >> 4;       // 0 or 1

    v8f acc[2][2];
    #pragma unroll
    for (int tm = 0; tm < 2; ++tm)
        #pragma unroll
        for (int tn = 0; tn < 2; ++tn)
            #pragma unroll
            for (int r = 0; r < 8; ++r) acc[tm][tn][r] = 0.0f;

    // prologue: stage first tile into buffer 0
    stage_tiles_async<LDA, LDB>(A, Bm, K, m0, n0, 0, tid, As[0], Bs[0]);

    int buf = 0;
    for (int k0 = 0; k0 < K; k0 += BK) {
        const bool hasNext = (k0 + BK) < K;
        if (hasNext) {
            // issue next stage into the other buffer, then retire only the
            // current stage's 6 wave-level async copies (in-order completion)
            stage_tiles_async<LDA, LDB>(A, Bm, K, m0, n0, k0 + BK, tid,
                                        As[buf ^ 1], Bs[buf ^ 1]);
            if (k0 + 2 * BK < K) {  // warm L2 for the stage after next
                __builtin_prefetch(A + (size_t)(m0 + (tid >> 3)) * K + k0 + 2 * BK, 0, 1);
                __builtin_prefetch(Bm + (size_t)(n0 + (tid >> 3)) * K + k0 + 2 * BK, 0, 1);
            }
            asm volatile("s_wait_asynccnt 0x6" ::: "memory");
        } else {
            asm volatile("s_wait_asynccnt 0x0" ::: "memory");
        }
        __syncthreads();

        const float* cAs = As[buf];
        const float* cBs = Bs[buf];

        // ---- 4 WMMA k-steps of depth 4 ----
        #pragma unroll
        for (int kk = 0; kk < BK; kk += 4) {
            const int kb = kk + 2 * laneHalf;  // K pair base for this lane half
            v2f afr[2], bfr[2];
            #pragma unroll
            for (int tm = 0; tm < 2; ++tm) {
                const int row = wm * 32 + tm * 16 + lane16;
                const float2 t2 = *(const float2*)&cAs[row * LDA + kb];
                afr[tm][0] = t2.x; afr[tm][1] = t2.y;
            }
            #pragma unroll
            for (int tn = 0; tn < 2; ++tn) {
                const int col = wn * 32 + tn * 16 + lane16;
                const float2 t2 = *(const float2*)&cBs[col * LDB + kb];
                bfr[tn][0] = t2.x; bfr[tn][1] = t2.y;
            }
            #pragma unroll
            for (int tm = 0; tm < 2; ++tm)
                #pragma unroll
                for (int tn = 0; tn < 2; ++tn)
                    acc[tm][tn] = __builtin_amdgcn_wmma_f32_16x16x4_f32(
                        false, afr[tm], false, bfr[tn],
                        (short)0, acc[tm][tn], false, false);
        }
        __syncthreads();   // all waves done reading before buffer is re-filled
        buf ^= 1;
    }

    // ---- epilogue: bias + activation, store ----
    #pragma unroll
    for (int tm = 0; tm < 2; ++tm) {
        #pragma unroll
        for (int tn = 0; tn < 2; ++tn) {
            const int nn = n0 + wn * 32 + tn * 16 + lane16;
            const float bv = bias[nn];
            #pragma unroll
            for (int r = 0; r < 8; ++r) {
                const int mm = m0 + wm * 32 + tm * 16 + laneHalf * 8 + r;
                float v = acc[tm][tn][r] + bv;
                if (EPI == 0)       v = expf(-(v * v));           // gaussian
                else                v = 1.0f / (1.0f + expf(-v)); // sigmoid
                C[(size_t)mm * N + nn] = v;
            }
        }
    }
}

// ---------------------------------------------------------------------------
// Block-wide sum reduction (256 threads)
// ---------------------------------------------------------------------------
__device__ __forceinline__ float blockReduceSum(float v, float* sdata)
{
    const int t = threadIdx.x;
    sdata[t] = v;
    __syncthreads();
    #pragma unroll
    for (int s = 128; s > 0; s >>= 1) {
        if (t < s) sdata[t] += sdata[t + s];
        __syncthreads();
    }
    const float r = sdata[0];
    __syncthreads();
    return r;
}

// ---------------------------------------------------------------------------
// Per-row combine: norms -> factor -> rotate -> mean -> cosine sim ->
// softmax over S -> superposition.  One 256-thread block per batch row.
// ---------------------------------------------------------------------------
__global__ __launch_bounds__(256)
void qsup_rowcombine(const float* __restrict__ O,      // B x (S*2H)
                     const float* __restrict__ phases, // S
                     float* __restrict__ supR,         // B x H
                     float* __restrict__ supI)         // B x H
{
    constexpr int HPT = HH / 256;   // 8 h-indices per thread
    __shared__ float sdata[256];

    const int b = blockIdx.x;
    const int t = threadIdx.x;
    const float* row = O + (size_t)b * N1;

    float cs[SS], sn[SS], factor[SS];
    #pragma unroll
    for (int s = 0; s < SS; ++s) {
        const float ph = phases[s];
        cs[s] = cosf(ph);
        sn[s] = sinf(ph);
    }

    // cache alpha/beta for this thread's h-lanes
    float aV[SS][HPT], bV[SS][HPT];
    #pragma unroll
    for (int s = 0; s < SS; ++s)
        #pragma unroll
        for (int i = 0; i < HPT; ++i) {
            const int h = t + i * 256;
            aV[s][i] = row[s * (2 * HH) + h];
            bV[s][i] = row[s * (2 * HH) + HH + h];
        }

    // partial-norm factor per stream
    #pragma unroll
    for (int s = 0; s < SS; ++s) {
        float loc = 0.f;
        #pragma unroll
        for (int i = 0; i < HPT; ++i)
            loc += aV[s][i] * aV[s][i] + bV[s][i] * bV[s][i];
        const float tot = blockReduceSum(loc, sdata) + EPSF;
        factor[s] = sqrtf(2.25f / tot);   // PARTIAL_NORM^2 = 1.5^2
    }

    // mean_real and per-stream wave norms
    float meanr[HPT];
    #pragma unroll
    for (int i = 0; i < HPT; ++i) meanr[i] = 0.f;
    float wnorm[SS];
    #pragma unroll
    for (int s = 0; s < SS; ++s) {
        float lsq = 0.f;
        #pragma unroll
        for (int i = 0; i < HPT; ++i) {
            const float re = factor[s] * (aV[s][i] * cs[s] - bV[s][i] * sn[s]);
            meanr[i] += re * (1.0f / SS);
            lsq += re * re;
        }
        wnorm[s] = sqrtf(blockReduceSum(lsq, sdata)) + EPSF;
    }

    float msq = 0.f;
    #pragma unroll
    for (int i = 0; i < HPT; ++i) msq += meanr[i] * meanr[i];
    const float meannorm = sqrtf(blockReduceSum(msq, sdata)) + EPSF;

    // cosine similarity per stream
    float csim[SS];
    #pragma unroll
    for (int s = 0; s < SS; ++s) {
        float ld = 0.f;
        #pragma unroll
        for (int i = 0; i < HPT; ++i) {
            const float re = factor[s] * (aV[s][i] * cs[s] - bV[s][i] * sn[s]);
            ld += re * meanr[i];
        }
        const float dot = blockReduceSum(ld, sdata);
        csim[s] = dot / (wnorm[s] * meannorm);
    }

    // softmax over S (every thread computes it; values are block-uniform)
    float mx = csim[0];
    #pragma unroll
    for (int s = 1; s < SS; ++s) mx = fmaxf(mx, csim[s]);
    float wsum = 0.f, wgt[SS];
    #pragma unroll
    for (int s = 0; s < SS; ++s) { wgt[s] = expf(csim[s] - mx); wsum += wgt[s]; }
    #pragma unroll
    for (int s = 0; s < SS; ++s) wgt[s] /= wsum;

    // superposition
    #pragma unroll
    for (int i = 0; i < HPT; ++i) {
        float sr = 0.f, si = 0.f;
        #pragma unroll
        for (int s = 0; s < SS; ++s) {
            const float re = factor[s] * (aV[s][i] * cs[s] - bV[s][i] * sn[s]);
            const float im = factor[s] * (aV[s][i] * sn[s] + bV[s][i] * cs[s]);
            sr += wgt[s] * re;
            si += wgt[s] * im;
        }
        const int h = t + i * 256;
        supR[(size_t)b * HH + h] = sr;
        supI[(size_t)b * HH + h] = si;
    }
}

// ---------------------------------------------------------------------------
// Elementwise helpers
// ---------------------------------------------------------------------------
__global__ void qsup_mag(const float* __restrict__ sr, const float* __restrict__ si,
                         float* __restrict__ mag, int n)
{
    const int i = blockIdx.x * blockDim.x + threadIdx.x;
    if (i < n) mag[i] = sqrtf(sr[i] * sr[i] + si[i] * si[i] + EPSF);
}

__global__ void qsup_apply_gate(float* __restrict__ sr, float* __restrict__ si,
                                const float* __restrict__ gate, int n)
{
    const int i = blockIdx.x * blockDim.x + threadIdx.x;
    if (i < n) { const float g = gate[i]; sr[i] *= g; si[i] *= g; }
}

__global__ void qsup_magsq(const float* __restrict__ sr, const float* __restrict__ si,
                           float* __restrict__ msq, int n)
{
    const int i = blockIdx.x * blockDim.x + threadIdx.x;
    if (i < n) msq[i] = sr[i] * sr[i] + si[i] * si[i];
}

// ---------------------------------------------------------------------------
// Top-64 of 2048 per row (iterative argmax, tie -> lowest index, matching
// stable descending top_k), then normalize to probs.
// ---------------------------------------------------------------------------
__global__ __launch_bounds__(256)
void qsup_topk(const float* __restrict__ msq,
               float* __restrict__ pvals, int* __restrict__ pinds)
{
    __shared__ float vals[HH];
    __shared__ float rv[256];
    __shared__ int   ri[256];
    __shared__ float topv[TOPK_K];
    __shared__ int   topi[TOPK_K];
    __shared__ float ssum;

    const int b = blockIdx.x;
    const int t = threadIdx.x;

    #pragma unroll
    for (int i = 0; i < HH / 256; ++i)
        vals[t + i * 256] = msq[(size_t)b * HH + t + i * 256];
    __syncthreads();

    for (int it = 0; it < TOPK_K; ++it) {
        float bv = -1.0f;
        int   bi = HH;
        #pragma unroll
        for (int i = 0; i < HH / 256; ++i) {
            const int idx = t + i * 256;
            const float v = vals[idx];
            if (v > bv || (v == bv && idx < bi)) { bv = v; bi = idx; }
        }
        rv[t] = bv; ri[t] = bi;
        __syncthreads();
        #pragma unroll
        for (int s = 128; s > 0; s >>= 1) {
            if (t < s) {
                const float ov = rv[t + s]; const int oi = ri[t + s];
                if (ov > rv[t] || (ov == rv[t] && oi < ri[t])) { rv[t] = ov; ri[t] = oi; }
            }
            __syncthreads();
        }
        if (t == 0) {
            topv[it] = rv[0];
            topi[it] = ri[0];
            vals[ri[0]] = -1.0f;   // remove (mag_sq >= 0, sentinel safe)
        }
        __syncthreads();
    }

    if (t == 0) {
        float s = 0.f;
        #pragma unroll
        for (int k = 0; k < TOPK_K; ++k) s += topv[k];
        ssum = s;
    }
    __syncthreads();

    if (t < TOPK_K) {
        pvals[(size_t)b * TOPK_K + t] = topv[t] / (ssum + EPSF);
        pinds[(size_t)b * TOPK_K + t] = topi[t];
    }
}

// ---------------------------------------------------------------------------
// Sparse classifier: out[b,c] = bc[c] + sum_k p[k] * Wc[c, h[k]]
// ---------------------------------------------------------------------------
__global__ __launch_bounds__(256)
void qsup_classify(const float* __restrict__ pvals, const int* __restrict__ pinds,
                   const float* __restrict__ Wc, const float* __restrict__ bc,
                   float* __restrict__ out)
{
    __shared__ float p[TOPK_K];
    __shared__ int   hh[TOPK_K];
    const int b = blockIdx.x;
    const int t = threadIdx.x;
    if (t < TOPK_K) {
        p[t]  = pvals[(size_t)b * TOPK_K + t];
        hh[t] = pinds[(size_t)b * TOPK_K + t];
    }
    __syncthreads();
    for (int c = t; c < CC; c += 256) {
        float acc = bc[c];
        const float* wrow = Wc + (size_t)c * HH;
        #pragma unroll 8
        for (int k = 0; k < TOPK_K; ++k)
            acc += p[k] * wrow[hh[k]];
        out[(size_t)b * CC + c] = acc;
    }
}

// ---------------------------------------------------------------------------
// Launch
// ---------------------------------------------------------------------------
extern "C" void kernel_launch(void* const* d_in, const int* in_sizes, int n_in,
                              void* d_out, int out_size, void* d_ws, size_t ws_size,
                              hipStream_t stream)
{
    const float* x      = (const float*)d_in[0];  // B x D
    const float* Wnet   = (const float*)d_in[1];  // S x 2H x D (flat: N1 x D)
    const float* bnet   = (const float*)d_in[2];  // N1
    const float* phases = (const float*)d_in[3];  // S
    const float* Wg     = (const float*)d_in[4];  // H x H
    const float* bg     = (const float*)d_in[5];  // H
    const float* Wc     = (const float*)d_in[6];  // C x H
    const float* bc     = (const float*)d_in[7];  // C
    float* out = (float*)d_out;

    // workspace layout
    char* ws = (char*)d_ws;
    const size_t SZ_O  = (size_t)BB * N1 * sizeof(float);      // 512 MB
    const size_t SZ_BH = (size_t)BB * HH * sizeof(float);      // 64 MB each
    const size_t SZ_PK = (size_t)BB * TOPK_K * sizeof(float);  // 2 MB each
    float* O    = (float*)(ws);
    float* supR = (float*)(ws + SZ_O);
    float* supI = (float*)(ws + SZ_O + SZ_BH);
    float* mag  = (float*)(ws + SZ_O + 2 * SZ_BH);             // also mag_sq
    float* gate = (float*)(ws + SZ_O + 3 * SZ_BH);
    float* pv   = (float*)(ws + SZ_O + 4 * SZ_BH);
    int*   pi   = (int*)  (ws + SZ_O + 4 * SZ_BH + SZ_PK);

    // 1) z = x @ Wnet^T + bnet ; O = exp(-z^2)     (WMMA fp32, async pipeline)
    qsup_gemm_nt_wmma<0><<<dim3(N1 / 128, BB / 64), 256, 0, stream>>>(
        x, Wnet, bnet, O, BB, N1, DD);

    // 2) per-row combine -> sup_real / sup_imag
    qsup_rowcombine<<<BB, 256, 0, stream>>>(O, phases, supR, supI);

    // 3) self-modulation x2: mag -> gate = sigmoid(mag @ Wg^T + bg) -> apply
    const int nBH = BB * HH;
    for (int step = 0; step < 2; ++step) {
        qsup_mag<<<(nBH + 255) / 256, 256, 0, stream>>>(supR, supI, mag, nBH);
        qsup_gemm_nt_wmma<1><<<dim3(HH / 128, BB / 64), 256, 0, stream>>>(
            mag, Wg, bg, gate, BB, HH, HH);
        qsup_apply_gate<<<(nBH + 255) / 256, 256, 0, stream>>>(supR, supI, gate, nBH);
    }

    // 4) mag_sq, top-64 -> probs
    qsup_magsq<<<(nBH + 255) / 256, 256, 0, stream>>>(supR, supI, mag, nBH);
    qsup_topk<<<BB, 256, 0, stream>>>(mag, pv, pi);

    // 5) sparse classifier -> out
    qsup_classify<<<BB, 256, 0, stream>>>(pv, pi, Wc, bc, out);
}